// GATv2_89696097010098
// MI455X (gfx1250) — compile-verified
//
#include <hip/hip_runtime.h>
#include <hip/hip_bf16.h>

typedef __attribute__((ext_vector_type(2))) float v2f;
typedef __attribute__((ext_vector_type(8))) float v8f;

// ---------------------------------------------------------------------------
// FP32 WMMA GEMM: C[M,NC] = A[M,K] @ B[K,NC]   (K, NC compile-time)
// Each wave32 computes a 16 x (16*NT) strip: one A fragment per k-step feeds
// NT independent V_WMMA_F32_16X16X4_F32 accumulator chains. Loads for step
// k+4 are issued before the WMMAs of step k (software double-buffering) so
// the waits become partial instead of full drains.
// A 16x4 layout: lanes 0-15 -> M=lane, v0=K0,v1=K1; lanes 16-31 -> v0=K2,v1=K3.
// B 4x16 layout: v0 row K=2*hi, v1 row K=2*hi+1, column N = lane&15.
// C/D: VGPR r holds M = r + 8*(lane>=16), N = lane&15.
// ---------------------------------------------------------------------------
template <int NT, int K, int NC>
__global__ __launch_bounds__(256)
void gemm_f32_wmma(const float* __restrict__ A, const float* __restrict__ B,
                   float* __restrict__ C, int M) {
  const int lane = threadIdx.x & 31;
  const int wid  = blockIdx.x * (blockDim.x >> 5) + (threadIdx.x >> 5);
  const int tilesM  = (M + 15) >> 4;
  constexpr int stripsN = (NC >> 4) / NT;
  if (wid >= tilesM * stripsN) return;              // wave-uniform: EXEC all-1s
  const int tm  = wid / stripsN;
  const int tn0 = (wid - tm * stripsN) * NT;
  const int hi = lane >> 4;                         // half-wave K-pair select
  const int lm = lane & 15;
  int arow = tm * 16 + lm; if (arow >= M) arow = M - 1;
  const int col0 = tn0 * 16 + lm;
  const float* __restrict__ Ar = A + (long)arow * K + 2 * hi;
  const float* __restrict__ Bc = B + (long)(2 * hi) * NC + col0;

  v8f c[NT];
#pragma unroll
  for (int t = 0; t < NT; ++t) c[t] = (v8f){};

  auto loadA = [&](int k) {
    v2f a; a.x = Ar[k]; a.y = Ar[k + 1]; return a;
  };
  auto loadB = [&](int k, v2f* b) {
    const float* __restrict__ B0 = Bc + (long)k * NC;
#pragma unroll
    for (int t = 0; t < NT; ++t) { b[t].x = B0[16 * t]; b[t].y = B0[NC + 16 * t]; }
  };

  v2f a_cur = loadA(0);
  v2f b_cur[NT];
  loadB(0, b_cur);
#pragma unroll 4
  for (int k = 0; k < K - 4; k += 4) {
    v2f a_nxt = loadA(k + 4);        // prefetch next step before WMMAs
    v2f b_nxt[NT];
    loadB(k + 4, b_nxt);
#pragma unroll
    for (int t = 0; t < NT; ++t)
      c[t] = __builtin_amdgcn_wmma_f32_16x16x4_f32(false, a_cur, false, b_cur[t],
                                                   (short)0, c[t], false, false);
    a_cur = a_nxt;
#pragma unroll
    for (int t = 0; t < NT; ++t) b_cur[t] = b_nxt[t];
  }
#pragma unroll
  for (int t = 0; t < NT; ++t)
    c[t] = __builtin_amdgcn_wmma_f32_16x16x4_f32(false, a_cur, false, b_cur[t],
                                                 (short)0, c[t], false, false);

#pragma unroll
  for (int t = 0; t < NT; ++t) {
#pragma unroll
    for (int r = 0; r < 8; ++r) {
      const int orow = tm * 16 + r + 8 * hi;
      if (orow < M) C[(long)orow * NC + col0 + 16 * t] = c[t][r];
    }
  }
}

// ---------------------------------------------------------------------------
// Per-layer init: zero aggregation buffer (optional), m = -inf bits, denom = 0.
// ---------------------------------------------------------------------------
__global__ void init_layer(float* out, unsigned* m, float* denom,
                           long outElems, int nh) {
  const long t = (long)blockIdx.x * blockDim.x + threadIdx.x;
  const long stride = (long)gridDim.x * blockDim.x;
  for (long i = t; i < outElems; i += stride) out[i] = 0.0f;
  for (long i = t; i < nh; i += stride) { m[i] = 0xFF800000u; denom[i] = 0.0f; }
}

__device__ __forceinline__ void atomic_max_f32(unsigned* p, float v) {
  if (v >= 0.0f) atomicMax((int*)p, __float_as_int(v));
  else           atomicMin(p, __float_as_uint(v));
}

// ---------------------------------------------------------------------------
// Edge logits + fused segment max: one wave per edge.
// s[e,h] = sum_d leakyrelu(f[src][h,d]+f[dst][h,d], 0.2) * attn[h,d]
// Lane 0 also folds s[e,h] into m[dst,h] via float-as-int atomic max.
// ---------------------------------------------------------------------------
template <int ROW, int JPH>
__global__ __launch_bounds__(256)
void edge_logits(const float* __restrict__ f, const int* __restrict__ src,
                 const int* __restrict__ dst, const float* __restrict__ attn,
                 float* __restrict__ s, unsigned* __restrict__ m, int E) {
  const int lane = threadIdx.x & 31;
  const int wid  = blockIdx.x * (blockDim.x >> 5) + (threadIdx.x >> 5);
  if (wid >= E) return;
  const long srow = (long)src[wid] * ROW;
  const long dn   = (long)dst[wid];
  const long drow = dn * ROW;
  constexpr int ITER = ROW / 32;
  constexpr int H = ITER / JPH;
  float acc[H];
#pragma unroll
  for (int h = 0; h < H; ++h) acc[h] = 0.0f;
#pragma unroll
  for (int j = 0; j < ITER; ++j) {
    const int d = j * 32 + lane;
    float v = f[srow + d] + f[drow + d];
    v = v > 0.0f ? v : 0.2f * v;
    acc[j / JPH] += v * attn[d];
  }
#pragma unroll
  for (int h = 0; h < H; ++h) {
    float v = acc[h];
#pragma unroll
    for (int off = 16; off > 0; off >>= 1) v += __shfl_xor(v, off, 32);
    if (lane == 0) {
      s[(long)wid * H + h] = v;
      atomic_max_f32(m + dn * H + h, v);
    }
  }
}

// ---------------------------------------------------------------------------
// exp(s - m[dst]) in-place, accumulate denominators.
// ---------------------------------------------------------------------------
__global__ void seg_exp(float* __restrict__ s, const int* __restrict__ dst,
                        const unsigned* __restrict__ m, float* __restrict__ denom,
                        long EH, int H) {
  const long t = (long)blockIdx.x * blockDim.x + threadIdx.x;
  if (t >= EH) return;
  const long e = t / H;
  const int  h = (int)(t - e * H);
  const long idx = (long)dst[e] * H + h;
  const float ex = __expf(s[t] - __uint_as_float(m[idx]));
  s[t] = ex;
  atomicAdd(denom + idx, ex);
}

// ---------------------------------------------------------------------------
// Message aggregation: out[dst] += f[src] * alpha, one wave per edge.
// ---------------------------------------------------------------------------
template <int ROW, int JPH>
__global__ __launch_bounds__(256)
void aggregate(const float* __restrict__ f, const float* __restrict__ ex,
               const float* __restrict__ denom, const int* __restrict__ src,
               const int* __restrict__ dst, float* __restrict__ out, int E) {
  const int lane = threadIdx.x & 31;
  const int wid  = blockIdx.x * (blockDim.x >> 5) + (threadIdx.x >> 5);
  if (wid >= E) return;
  constexpr int ITER = ROW / 32;
  constexpr int H = ITER / JPH;
  const long srow = (long)src[wid] * ROW;
  const long dn   = (long)dst[wid];
  float al[H];
#pragma unroll
  for (int h = 0; h < H; ++h) al[h] = ex[(long)wid * H + h] / denom[dn * H + h];
#pragma unroll
  for (int j = 0; j < ITER; ++j) {
    const int d = j * 32 + lane;
    atomicAdd(out + dn * ROW + d, f[srow + d] * al[j / JPH]);
  }
}

__global__ void elu_inplace(float* __restrict__ x, long n) {
  const long t = (long)blockIdx.x * blockDim.x + threadIdx.x;
  if (t >= n) return;
  const float v = x[t];
  x[t] = v > 0.0f ? v : expm1f(v);
}

// Mean over 4 heads of 32 classes each: out[n,c] = mean_h in[n, h*32+c]
__global__ void head_mean(const float* __restrict__ in, float* __restrict__ out,
                          int N) {
  const int t = blockIdx.x * blockDim.x + threadIdx.x;
  if (t >= N * 32) return;
  const int n = t >> 5, c = t & 31;
  const float* r = in + (long)n * 128 + c;
  out[t] = 0.25f * (r[0] + r[32] + r[64] + r[96]);
}

// ---------------------------------------------------------------------------
static inline int cdiv_l(long a, long b) { return (int)((a + b - 1) / b); }

extern "C" void kernel_launch(void* const* d_in, const int* in_sizes, int n_in,
                              void* d_out, int out_size, void* d_ws, size_t ws_size,
                              hipStream_t stream) {
  const float* x     = (const float*)d_in[0];
  const float* W0    = (const float*)d_in[1];
  const float* a0    = (const float*)d_in[2];   // [4,64] flat = 256
  const float* W1    = (const float*)d_in[3];
  const float* a1    = (const float*)d_in[4];   // [4,32] flat = 128
  const float* Wres1 = (const float*)d_in[5];
  const int*   src   = (const int*)d_in[6];
  const int*   dst   = (const int*)d_in[7];

  const int N = in_sizes[0] / 256;
  const int E = in_sizes[6];
  const int H = 4;

  // Workspace carve-out
  float*    f0    = (float*)d_ws;                 // [N,256] layer-0 projection
  float*    h     = f0 + (long)N * 256;           // [N,256] layer-0 agg -> ELU
  float*    f1    = h  + (long)N * 256;           // [N,128] layer-1 projection
  float*    o1    = f1 + (long)N * 128;           // [N,128] residual + agg
  float*    s     = o1 + (long)N * 128;           // [E,4] logits / exp
  unsigned* m     = (unsigned*)(s + (long)E * H); // [N,4] segment max (bits)
  float*    denom = (float*)(m + (long)N * H);    // [N,4]
  const size_t need =
      ((size_t)N * 256 * 2 + (size_t)N * 128 * 2 + (size_t)E * H + (size_t)N * H * 2)
      * sizeof(float);
  if (need > ws_size) return;

  const int BT = 256;
  const long EH = (long)E * H;
  const int egrid = cdiv_l((long)E * 32, BT);     // one wave per edge

  // ------------------ Layer 0 ------------------
  {
    const int strips = ((N + 15) / 16) * (256 / 16 / 4);   // 16x64 per wave
    gemm_f32_wmma<4, 256, 256><<<cdiv_l(strips, 8), BT, 0, stream>>>(x, W0, f0, N);
  }
  init_layer<<<cdiv_l((long)N * 256, BT), BT, 0, stream>>>(h, m, denom,
                                                           (long)N * 256, N * H);
  edge_logits<256, 2><<<egrid, BT, 0, stream>>>(f0, src, dst, a0, s, m, E);
  seg_exp<<<cdiv_l(EH, BT), BT, 0, stream>>>(s, dst, m, denom, EH, H);
  aggregate<256, 2><<<egrid, BT, 0, stream>>>(f0, s, denom, src, dst, h, E);
  elu_inplace<<<cdiv_l((long)N * 256, BT), BT, 0, stream>>>(h, (long)N * 256);

  // ------------------ Layer 1 ------------------
  {
    const int strips = ((N + 15) / 16) * (128 / 16 / 4);
    gemm_f32_wmma<4, 256, 128><<<cdiv_l(strips, 8), BT, 0, stream>>>(h, W1, f1, N);
    gemm_f32_wmma<4, 256, 128><<<cdiv_l(strips, 8), BT, 0, stream>>>(h, Wres1, o1, N);
  }
  init_layer<<<cdiv_l((long)N * H, BT), BT, 0, stream>>>(nullptr, m, denom,
                                                         0, N * H);
  edge_logits<128, 1><<<egrid, BT, 0, stream>>>(f1, src, dst, a1, s, m, E);
  seg_exp<<<cdiv_l(EH, BT), BT, 0, stream>>>(s, dst, m, denom, EH, H);
  aggregate<128, 1><<<egrid, BT, 0, stream>>>(f1, s, denom, src, dst, o1, E);

  head_mean<<<cdiv_l((long)N * 32, BT), BT, 0, stream>>>(o1, (float*)d_out, N);
}